// TrainSpeakerSimilarity_7387343749203
// MI455X (gfx1250) — compile-verified
//
#include <hip/hip_runtime.h>
#include <hip/hip_bf16.h>
#include <math.h>

#define NSPK 256
#define MUTT 64
#define DDIM 1024
#define KCHUNK 64
#define LDSTR 68   // stride % 64 == 4 -> b64 frag reads hit all 64 banks conflict-free

typedef __attribute__((ext_vector_type(2))) float v2f;
typedef __attribute__((ext_vector_type(8))) float v8f;

// ---------------------------------------------------------------------------
// Kernel A: one block per speaker j.
//   S[j][d] = sum_m E[j*M+m][d];  e2[r] = ||E[r]||^2;  S2[j] = ||S_j||^2
//   cs[j]   = (1/M) * rsqrt(max(S2/M^2, eps))
// ---------------------------------------------------------------------------
__global__ __launch_bounds__(256) void spk_centroid_kernel(
    const float* __restrict__ E, float* __restrict__ S,
    float* __restrict__ e2, float* __restrict__ S2a, float* __restrict__ csa)
{
  const int j    = blockIdx.x;
  const int t    = threadIdx.x;
  const int lane = t & 31;
  const int wid  = t >> 5;
  __shared__ float red[8];

  float c0 = 0.f, c1 = 0.f, c2 = 0.f, c3 = 0.f;
  const float* base = E + (size_t)j * MUTT * DDIM;
  for (int m = 0; m < MUTT; ++m) {
    const float* row = base + (size_t)m * DDIM;
    float v0 = row[t], v1 = row[t + 256], v2 = row[t + 512], v3 = row[t + 768];
    c0 += v0; c1 += v1; c2 += v2; c3 += v3;
    float p = v0 * v0 + v1 * v1 + v2 * v2 + v3 * v3;
#pragma unroll
    for (int off = 16; off > 0; off >>= 1) p += __shfl_xor(p, off, 32);
    if (lane == 0) red[wid] = p;
    __syncthreads();
    if (t == 0) {
      e2[j * MUTT + m] =
          red[0] + red[1] + red[2] + red[3] + red[4] + red[5] + red[6] + red[7];
    }
    __syncthreads();
  }

  float* Sj = S + (size_t)j * DDIM;
  Sj[t] = c0; Sj[t + 256] = c1; Sj[t + 512] = c2; Sj[t + 768] = c3;

  float q = c0 * c0 + c1 * c1 + c2 * c2 + c3 * c3;
#pragma unroll
  for (int off = 16; off > 0; off >>= 1) q += __shfl_xor(q, off, 32);
  if (lane == 0) red[wid] = q;
  __syncthreads();
  if (t == 0) {
    float s2 = red[0] + red[1] + red[2] + red[3] + red[4] + red[5] + red[6] + red[7];
    S2a[j] = s2;
    const float invM = 1.0f / (float)MUTT;
    float m2 = s2 * invM * invM;
    csa[j] = invM * rsqrtf(fmaxf(m2, 1e-12f));
  }
}

// ---------------------------------------------------------------------------
// Kernel B: G = E (16384x1024) * S^T (1024x256) via v_wmma_f32_16x16x4_f32.
// Block tile: 64 rows (one speaker) x 128 cols; 8 waves as 2x4; each wave
// holds a 2x2 grid of 16x16 accumulators (4 WMMAs per pair of A/B frag loads,
// all operands served from LDS).  Fused epilogue:
//   out[r][i] = w * ( i==j ? (G - e2[r]) * rsqrt(max(S2[j]-2G+e2[r], eps))
//                          :  G * cs[i] ) + b
// ---------------------------------------------------------------------------
__global__ __launch_bounds__(256) void spk_sim_gemm_kernel(
    const float* __restrict__ E, const float* __restrict__ S,
    const float* __restrict__ e2, const float* __restrict__ S2a,
    const float* __restrict__ csa, const float* __restrict__ wv,
    const float* __restrict__ bv, float* __restrict__ out)
{
  __shared__ float As[64 * LDSTR];    // 17.0 KB
  __shared__ float Bs[128 * LDSTR];   // 34.0 KB

  const int t    = threadIdx.x;
  const int wave = t >> 5;
  const int lane = t & 31;
  const int n    = lane & 15;     // column within 16-wide tile
  const int h    = lane >> 4;     // half-wave -> K+2 per ISA A/B layout
  const int wr   = wave >> 2;     // 0..1  row group
  const int wc   = wave & 3;      // 0..3  col group

  const int rowbase = blockIdx.y * 64;        // one speaker per block
  const int j       = blockIdx.y;
  const int colbase = blockIdx.x * 128;

  const int rb0 = wr * 32, rb1 = rb0 + 16;    // local row-tile origins
  const int cb0 = wc * 32, cb1 = cb0 + 16;    // local col-tile origins

  v8f acc00 = {}; v8f acc01 = {}; v8f acc10 = {}; v8f acc11 = {};

  // cooperative staging maps (fully coalesced b128, 16B-aligned LDS stores)
  const int ar = t >> 2, ac = (t & 3) * 16;   // A: 4 thr/row, 16 floats each
  const int br = t >> 1, bc = (t & 1) * 32;   // B: 2 thr/row, 32 floats each
  const float* aptr = E + (size_t)(rowbase + ar) * DDIM + ac;
  const float* bptr = S + (size_t)(colbase + br) * DDIM + bc;

  for (int kb = 0; kb < DDIM; kb += KCHUNK) {
    const float4* asrc = (const float4*)(aptr + kb);
    float4 a0 = asrc[0], a1 = asrc[1], a2 = asrc[2], a3 = asrc[3];
    const float4* bsrc = (const float4*)(bptr + kb);
    float4 b0 = bsrc[0], b1 = bsrc[1], b2 = bsrc[2], b3 = bsrc[3];
    float4 b4 = bsrc[4], b5 = bsrc[5], b6 = bsrc[6], b7 = bsrc[7];

    float4* adst = (float4*)&As[ar * LDSTR + ac];
    adst[0] = a0; adst[1] = a1; adst[2] = a2; adst[3] = a3;
    float4* bdst = (float4*)&Bs[br * LDSTR + bc];
    bdst[0] = b0; bdst[1] = b1; bdst[2] = b2; bdst[3] = b3;
    bdst[4] = b4; bdst[5] = b5; bdst[6] = b6; bdst[7] = b7;

    if (kb + KCHUNK < DDIM) {                 // warm L2/L0 for next chunk
      __builtin_prefetch(aptr + kb + KCHUNK, 0, 0);
      __builtin_prefetch(bptr + kb + KCHUNK, 0, 0);
    }
    __syncthreads();

#pragma unroll
    for (int kk = 0; kk < KCHUNK; kk += 4) {
      v2f a0f = *(const v2f*)&As[(rb0 + n) * LDSTR + kk + 2 * h];
      v2f a1f = *(const v2f*)&As[(rb1 + n) * LDSTR + kk + 2 * h];
      v2f b0f = *(const v2f*)&Bs[(cb0 + n) * LDSTR + kk + 2 * h];
      v2f b1f = *(const v2f*)&Bs[(cb1 + n) * LDSTR + kk + 2 * h];
      acc00 = __builtin_amdgcn_wmma_f32_16x16x4_f32(false, a0f, false, b0f,
                                                    (short)0, acc00, false, false);
      acc01 = __builtin_amdgcn_wmma_f32_16x16x4_f32(false, a0f, false, b1f,
                                                    (short)0, acc01, false, false);
      acc10 = __builtin_amdgcn_wmma_f32_16x16x4_f32(false, a1f, false, b0f,
                                                    (short)0, acc10, false, false);
      acc11 = __builtin_amdgcn_wmma_f32_16x16x4_f32(false, a1f, false, b1f,
                                                    (short)0, acc11, false, false);
    }
    __syncthreads();
  }

  // ---- fused epilogue ----
  const float w   = wv[0];
  const float b   = bv[0];
  const float s2j = S2a[j];
  const int   i0  = colbase + cb0 + n;
  const int   i1  = colbase + cb1 + n;
  const float cs0 = csa[i0];
  const float cs1 = csa[i1];

  auto emit = [&](float g, int r, int ig, float csv, float er) {
    float val = (ig == j)
                    ? (g - er) * rsqrtf(fmaxf(s2j - 2.0f * g + er, 1e-12f))
                    : g * csv;
    out[(size_t)r * NSPK + ig] = w * val + b;
  };

#pragma unroll
  for (int v = 0; v < 8; ++v) {
    const int m  = v + 8 * h;               // C/D layout: VGPR v -> row v (+8 hi)
    const int r0 = rowbase + rb0 + m;
    const int r1 = rowbase + rb1 + m;
    const float er0 = e2[r0];
    const float er1 = e2[r1];
    emit(acc00[v], r0, i0, cs0, er0);
    emit(acc01[v], r0, i1, cs1, er0);
    emit(acc10[v], r1, i0, cs0, er1);
    emit(acc11[v], r1, i1, cs1, er1);
  }
}

// ---------------------------------------------------------------------------
extern "C" void kernel_launch(void* const* d_in, const int* in_sizes, int n_in,
                              void* d_out, int out_size, void* d_ws, size_t ws_size,
                              hipStream_t stream) {
  (void)in_sizes; (void)n_in; (void)out_size; (void)ws_size;
  const float* E  = (const float*)d_in[0];   // (N*M, D) f32
  const float* wv = (const float*)d_in[1];   // scalar
  const float* bv = (const float*)d_in[2];   // scalar
  float* out = (float*)d_out;                // (N*M, N) f32

  // workspace: S (256*1024) | e2 (16384) | S2 (256) | cs (256)  ~= 1.12 MB
  float* S  = (float*)d_ws;
  float* e2 = S + (size_t)NSPK * DDIM;
  float* S2 = e2 + (size_t)NSPK * MUTT;
  float* cs = S2 + NSPK;

  spk_centroid_kernel<<<NSPK, 256, 0, stream>>>(E, S, e2, S2, cs);
  dim3 grid(NSPK / 128, (NSPK * MUTT) / 64);  // (2 col halves, 256 row strips)
  spk_sim_gemm_kernel<<<grid, 256, 0, stream>>>(E, S, e2, S2, cs, wv, bv, out);
}